// SphericalExpansion_558345748600
// MI455X (gfx1250) — compile-verified
//
#include <hip/hip_runtime.h>
#include <hip/hip_bf16.h>
#include <math.h>

// ---------------------------------------------------------------------------
// SphericalExpansion on gfx1250 (MI455X).
// out[a, p*128 + lm*8 + n] = sum_{e: center[e]==a} Y_lm(e) * rb_n(e) * pw_p(e)
// Strategy: counting-sort edges by center, then per-center GEMM
//   D[16x32] = Y[16 x K] * W[K x 32]   via V_WMMA_F32_16X16X4_F32 (K chunks of 4)
// eliminating all output atomics (naive path = 327M L2 atomics, ~1000x over
// the 23.3 TB/s memory roofline; this path moves ~46 MB total).
// Wave-uniform control flow is forced via readfirstlane so EXEC is provably
// all-1s at every WMMA (ISA requirement).
// ---------------------------------------------------------------------------

typedef __attribute__((ext_vector_type(2))) float v2f;
typedef __attribute__((ext_vector_type(8))) float v8f;

#define WPB   4      // waves per block (wave32)
#define YSTR  17     // LDS row stride (dwords) for Y  (17 coprime to 64 banks)
#define WSTR  49     // LDS row stride (dwords) for W  (49 coprime to 64 banks)

// ---------------- counting sort helpers ----------------

__global__ void se_zero_i32(int* __restrict__ p, int n) {
    int i = blockIdx.x * blockDim.x + threadIdx.x;
    if (i < n) p[i] = 0;
}

__global__ void se_hist(const int* __restrict__ centers, int* __restrict__ counts, int E) {
    int i = blockIdx.x * blockDim.x + threadIdx.x;
    if (i < E) atomicAdd(&counts[centers[i] + 1], 1);   // shifted: inclusive scan -> exclusive offsets
}

// single-workgroup inclusive scan over n ints (n ~ 20001)
__global__ __launch_bounds__(1024) void se_scan(int* __restrict__ data, int n) {
    __shared__ int tmp[1024];
    __shared__ int carry;
    if (threadIdx.x == 0) carry = 0;
    __syncthreads();
    for (int base = 0; base < n; base += 1024) {
        int i = base + (int)threadIdx.x;
        int v = (i < n) ? data[i] : 0;
        tmp[threadIdx.x] = v;
        __syncthreads();
        for (int off = 1; off < 1024; off <<= 1) {
            int t = (threadIdx.x >= (unsigned)off) ? tmp[threadIdx.x - off] : 0;
            __syncthreads();
            tmp[threadIdx.x] += t;
            __syncthreads();
        }
        int inc = tmp[threadIdx.x] + carry;
        if (i < n) data[i] = inc;
        __syncthreads();
        if (threadIdx.x == 1023) carry = inc;
        __syncthreads();
    }
}

__global__ void se_cursor_init(const int* __restrict__ offsets, int* __restrict__ cursors, int A) {
    int i = blockIdx.x * blockDim.x + threadIdx.x;
    if (i < A) cursors[i] = offsets[i];
}

__global__ void se_scatter(const int* __restrict__ centers, int* __restrict__ cursors,
                           int* __restrict__ sorted, int E) {
    int i = blockIdx.x * blockDim.x + threadIdx.x;
    if (i < E) {
        int pos = atomicAdd(&cursors[centers[i]], 1);
        sorted[pos] = i;
    }
}

// ---------------- main per-center WMMA kernel ----------------

__global__ __launch_bounds__(WPB * 32) void se_main(
    const float* __restrict__ positions,     // [A][3]
    const float* __restrict__ comb_weight,   // [4][4] row-major [p][s]
    const int*   __restrict__ neighbors,     // [E]
    const int*   __restrict__ species_idx,   // [A]
    const int*   __restrict__ offsets,       // [A+1] exclusive starts
    const int*   __restrict__ sorted,        // [E] edge ids sorted by center
    float*       __restrict__ out,           // [A][512]
    int A)
{
    __shared__ float Yl[WPB][32 * YSTR];
    __shared__ float Wl[WPB][32 * WSTR];

    const int lane = threadIdx.x & 31;
    // Force wave-uniform values into SGPRs: guarantees scalar branches
    // (s_cbranch) around the WMMA loop, so EXEC stays all-1s as required.
    const int wv = __builtin_amdgcn_readfirstlane((int)(threadIdx.x >> 5));
    const int a  = (int)blockIdx.x * WPB + wv;
    if (a >= A) return;

    const int start = __builtin_amdgcn_readfirstlane(offsets[a]);
    const int k     = __builtin_amdgcn_readfirstlane(offsets[a + 1]) - start;

    const float cx = positions[3 * a + 0];
    const float cy = positions[3 * a + 1];
    const float cz = positions[3 * a + 2];

    v8f acc0 = {};   // D tile, W-columns 0..15  (p = 0,1)
    v8f acc1 = {};   // D tile, W-columns 16..31 (p = 2,3)

    float* Ys = Yl[wv];
    float* Ws = Wl[wv];

    const int mrow = lane & 15;
    const int hlf  = lane >> 4;               // 0: K columns {0,1}; 1: K columns {2,3}

    for (int base = 0; base < k; base += 32) {
        const int  rem   = k - base;
        const bool valid = lane < rem;
        const int  ei    = base + (valid ? lane : 0);
        const int  e     = sorted[start + ei];
        const int  nb    = neighbors[e];

        float x = positions[3 * nb + 0] - cx;
        float y = positions[3 * nb + 1] - cy;
        float z = positions[3 * nb + 2] - cz;
        const float r     = sqrtf(x * x + y * y + z * z + 1e-12f);
        const float inv_r = 1.0f / r;
        x *= inv_r; y *= inv_r; z *= inv_r;

        // radial: fc = 0.5*(cos(pi r/Rc)+1)*(r<Rc);  rb[n] = fc*sin((n+1) pi r/Rc)/r
        const float theta = (3.14159265358979323846f / 5.0f) * r;
        const float s1 = sinf(theta);
        const float c1 = cosf(theta);
        const float fc = (r < 5.0f) ? (0.5f * (c1 + 1.0f)) : 0.0f;
        const float msk = valid ? (fc * inv_r) : 0.0f;   // tail masking flows through W

        float rb[8];
        {
            float sn = s1, cn = c1;
            rb[0] = msk * sn;
#pragma unroll
            for (int n = 1; n < 8; ++n) {
                const float sn1 = sn * c1 + cn * s1;
                const float cn1 = cn * c1 - sn * s1;
                sn = sn1; cn = cn1;
                rb[n] = msk * sn;
            }
        }

        const int sp = species_idx[nb];
        const float pw[4] = { comb_weight[0 * 4 + sp], comb_weight[1 * 4 + sp],
                              comb_weight[2 * 4 + sp], comb_weight[3 * 4 + sp] };

        // W column order w = p*8 + n  ->  coalesced output stores later
#pragma unroll
        for (int p = 0; p < 4; ++p)
#pragma unroll
            for (int n = 0; n < 8; ++n)
                Ws[lane * WSTR + p * 8 + n] = rb[n] * pw[p];

        // real spherical harmonics l=0..3 (16 components, l-major then m)
        const float xx = x * x, yy = y * y, zz = z * z;
        float Yv[16];
        Yv[0]  = 0.28209479177387814f;
        Yv[1]  = 0.4886025119029199f * y;
        Yv[2]  = 0.4886025119029199f * z;
        Yv[3]  = 0.4886025119029199f * x;
        Yv[4]  = 1.0925484305920792f * x * y;
        Yv[5]  = 1.0925484305920792f * y * z;
        Yv[6]  = 0.31539156525252005f * (3.0f * zz - 1.0f);
        Yv[7]  = 1.0925484305920792f * x * z;
        Yv[8]  = 0.5462742152960396f * (xx - yy);
        Yv[9]  = 0.5900435899266435f * y * (3.0f * xx - yy);
        Yv[10] = 2.890611442640554f  * x * y * z;
        Yv[11] = 0.4570457994644658f * y * (5.0f * zz - 1.0f);
        Yv[12] = 0.3731763325901154f * (5.0f * zz * z - 3.0f * z);
        Yv[13] = 0.4570457994644658f * x * (5.0f * zz - 1.0f);
        Yv[14] = 1.445305721320277f  * z * (xx - yy);
        Yv[15] = 0.5900435899266435f * x * (xx - 3.0f * yy);
#pragma unroll
        for (int i = 0; i < 16; ++i) Ys[lane * YSTR + i] = Yv[i];

        // consume the 32 staged edges as 8 WMMA K-chunks of 4
        // A (16x4 f32): VGPR0 = K{0|2}, VGPR1 = K{1|3}; lanes 0-15 -> K0/K1, 16-31 -> K2/K3
        // B (4x16 f32): same K striping; lane%16 selects N
#pragma unroll
        for (int c = 0; c < 8; ++c) {
            const int e0 = c * 4 + 2 * hlf;
            v2f av, b0v, b1v;
            av.x  = Ys[(e0 + 0) * YSTR + mrow];
            av.y  = Ys[(e0 + 1) * YSTR + mrow];
            b0v.x = Ws[(e0 + 0) * WSTR + mrow];
            b0v.y = Ws[(e0 + 1) * WSTR + mrow];
            b1v.x = Ws[(e0 + 0) * WSTR + 16 + mrow];
            b1v.y = Ws[(e0 + 1) * WSTR + 16 + mrow];
            acc0 = __builtin_amdgcn_wmma_f32_16x16x4_f32(
                false, av, false, b0v, (short)0, acc0, false, false);
            acc1 = __builtin_amdgcn_wmma_f32_16x16x4_f32(
                false, av, false, b1v, (short)0, acc1, false, false);
        }
    }

    // D 16x16 f32 layout: VGPR r, lane L -> M = r + 8*(L>>4), N = L&15
    // w = tile*16 + N with w = p*8 + n  ->  out[a, (w>>3)*128 + lm*8 + (w&7)]
    // lanes 0..7 / 8..15 each write 8 consecutive dwords per rI (coalesced).
    float* outa = out + (size_t)a * 512;
#pragma unroll
    for (int rI = 0; rI < 8; ++rI) {
        const int lm = rI + 8 * hlf;
        const int w0 = mrow;
        const int w1 = 16 + mrow;
        outa[(w0 >> 3) * 128 + lm * 8 + (w0 & 7)] = acc0[rI];
        outa[(w1 >> 3) * 128 + lm * 8 + (w1 & 7)] = acc1[rI];
    }
}

// ---------------- launcher ----------------

extern "C" void kernel_launch(void* const* d_in, const int* in_sizes, int n_in,
                              void* d_out, int out_size, void* d_ws, size_t ws_size,
                              hipStream_t stream) {
    const float* positions   = (const float*)d_in[0];
    const float* comb_weight = (const float*)d_in[1];
    const int*   centers     = (const int*)d_in[2];
    const int*   neighbors   = (const int*)d_in[3];
    const int*   species_idx = (const int*)d_in[4];
    float* out = (float*)d_out;

    const int A = in_sizes[4];      // species_idx has A elements
    const int E = in_sizes[2];      // centers has E elements

    // workspace: offsets[A+1] | cursors[A] | sorted[E]   (~2.7 MB)
    int* offsets = (int*)d_ws;
    int* cursors = offsets + (A + 1);
    int* sorted  = cursors + A;

    const int T = 256;
    se_zero_i32<<<(A + 1 + T - 1) / T, T, 0, stream>>>(offsets, A + 1);
    se_hist<<<(E + T - 1) / T, T, 0, stream>>>(centers, offsets, E);
    se_scan<<<1, 1024, 0, stream>>>(offsets, A + 1);
    se_cursor_init<<<(A + T - 1) / T, T, 0, stream>>>(offsets, cursors, A);
    se_scatter<<<(E + T - 1) / T, T, 0, stream>>>(centers, cursors, sorted, E);

    se_main<<<(A + WPB - 1) / WPB, WPB * 32, 0, stream>>>(
        positions, comb_weight, neighbors, species_idx, offsets, sorted, out, A);
}